// PMF_32796370272277
// MI455X (gfx1250) — compile-verified
//
#include <hip/hip_runtime.h>
#include <hip/hip_bf16.h>

// ---------------------------------------------------------------------------
// PMF loss + preds for MI455X (gfx1250, wave32, WMMA)
//
//   loss  = sum((R != 0) * (R - U V^T)^2) + 0.01*(sum U^2 + sum V^2)
//   preds = rowwise dot of U[u_idx, :4] and V[v_idx, :4]
//
// Bandwidth-bound on streaming R (576 MB, read once).  U V^T tiles are
// computed on the fly with V_WMMA_F32_16X16X4_F32 (two chained WMMAs for K=5).
// ---------------------------------------------------------------------------

#define NU 12000
#define NI 12000
#define LDIM 5
#define COL_CHUNKS 25              // 750 column tiles / 25 chunks
#define TILES_PER_CHUNK 30         // 30 tiles of 16 cols = 480 cols per chunk
#define ROW_STRIPS (NU / 16)       // 750
#define WAVES_TOTAL (ROW_STRIPS * COL_CHUNKS)   // 18750
#define WAVES_PER_BLOCK 6          // 18750 / 6 = 3125 blocks exactly
#define REDUCE_THREADS 256

typedef float v2f __attribute__((ext_vector_type(2)));
typedef float v8f __attribute__((ext_vector_type(8)));

// ---------------------------------------------------------------------------
// Kernel 1: masked squared-error over R, one 16-row strip x 480-col chunk per
// wave.  Per-wave partial sum -> ws[wave].
// ---------------------------------------------------------------------------
__global__ __launch_bounds__(32 * WAVES_PER_BLOCK)
void pmf_loss_kernel(const float* __restrict__ U,
                     const float* __restrict__ V,
                     const float* __restrict__ R,
                     float* __restrict__ ws)
{
    const int wave  = blockIdx.x * WAVES_PER_BLOCK + (threadIdx.x >> 5);
    const int lane  = threadIdx.x & 31;
    const int strip = wave / COL_CHUNKS;          // 0..749
    const int chunk = wave % COL_CHUNKS;          // 0..24

    const int  m0      = strip * 16;
    const int  n0_base = chunk * TILES_PER_CHUNK * 16;
    const int  l15     = lane & 15;
    const bool khi     = (lane >= 16);            // lanes 16..31 carry K=2,3

    // ---- A fragment: 16x4 slice of U rows m0..m0+15 (K = 0..3), plus a
    //      zero-padded fragment carrying K = 4.
    const float* Urow = U + (m0 + l15) * LDIM;
    v2f a1, a2;
    a1.x = Urow[khi ? 2 : 0];
    a1.y = Urow[khi ? 3 : 1];
    a2.x = khi ? 0.0f : Urow[4];
    a2.y = 0.0f;

    float acc = 0.0f;

    for (int t = 0; t < TILES_PER_CHUNK; ++t) {
        const int n0 = n0_base + t * 16;

        // ---- B fragment: 4x16 slice of V^T (columns n0..n0+15), same
        //      striping as A but indexed by column.
        const float* Vrow = V + (n0 + l15) * LDIM;
        v2f b1, b2;
        b1.x = Vrow[khi ? 2 : 0];
        b1.y = Vrow[khi ? 3 : 1];
        b2.x = khi ? 0.0f : Vrow[4];
        b2.y = 0.0f;

        // ---- P(16x16) = A * B  via two chained f32 WMMAs (K = 0..3, K = 4)
        v8f c = {0.f, 0.f, 0.f, 0.f, 0.f, 0.f, 0.f, 0.f};
        c = __builtin_amdgcn_wmma_f32_16x16x4_f32(
                false, a1, false, b1, (short)0, c, false, false);
        c = __builtin_amdgcn_wmma_f32_16x16x4_f32(
                false, a2, false, b2, (short)0, c, false, false);

        // ---- Stream the matching 16x16 tile of R and accumulate masked
        //      squared error.  C layout: VGPR i <-> row m0+i (lanes 0-15) or
        //      m0+8+i (lanes 16-31); lane&15 <-> column n0+l15.
        const size_t rbase = (size_t)(m0 + (khi ? 8 : 0)) * NI + (size_t)(n0 + l15);
        const float* Rp = R + rbase;

        // Prefetch the next tile's first row (gfx1250 global_prefetch_b8).
        if (t + 1 < TILES_PER_CHUNK)
            __builtin_prefetch(Rp + 16, 0, 1);

#pragma unroll
        for (int i = 0; i < 8; ++i) {
            const float r = Rp[(size_t)i * NI];
            const float e = r - c[i];
            acc += (r != 0.0f) ? (e * e) : 0.0f;
        }
    }

    // ---- wave32 reduction (fixed order -> deterministic)
#pragma unroll
    for (int off = 16; off >= 1; off >>= 1)
        acc += __shfl_xor(acc, off, 32);

    if (lane == 0)
        ws[wave] = acc;
}

// ---------------------------------------------------------------------------
// Kernel 2: deterministic final reduction + L2 regularization terms.
// Single block, fixed-order strided accumulation + shared-memory tree.
// ---------------------------------------------------------------------------
__global__ __launch_bounds__(REDUCE_THREADS)
void pmf_reduce_kernel(const float* __restrict__ ws,
                       const float* __restrict__ U,
                       const float* __restrict__ V,
                       float* __restrict__ out)
{
    __shared__ float sm[REDUCE_THREADS];
    const int tid = threadIdx.x;

    float s = 0.0f;
    for (int i = tid; i < WAVES_TOTAL; i += REDUCE_THREADS)
        s += ws[i];

    float reg = 0.0f;
    for (int i = tid; i < NU * LDIM; i += REDUCE_THREADS)
        reg += U[i] * U[i];
    for (int i = tid; i < NI * LDIM; i += REDUCE_THREADS)
        reg += V[i] * V[i];
    s += 0.01f * reg;   // LAMBDA_U == LAMBDA_V == 0.01

    sm[tid] = s;
    __syncthreads();
#pragma unroll
    for (int off = REDUCE_THREADS / 2; off >= 1; off >>= 1) {
        if (tid < off) sm[tid] += sm[tid + off];
        __syncthreads();
    }
    if (tid == 0) out[0] = sm[0];
}

// ---------------------------------------------------------------------------
// Kernel 3: gathered 4-dim dot products (preds).
// ---------------------------------------------------------------------------
__global__ __launch_bounds__(256)
void pmf_preds_kernel(const float* __restrict__ U,
                      const float* __restrict__ V,
                      const int* __restrict__ u_idx,
                      const int* __restrict__ v_idx,
                      float* __restrict__ preds,
                      int np)
{
    const int i = blockIdx.x * blockDim.x + threadIdx.x;
    if (i >= np) return;
    const float* ur = U + (size_t)u_idx[i] * LDIM;
    const float* vr = V + (size_t)v_idx[i] * LDIM;
    preds[i] = ur[0] * vr[0] + ur[1] * vr[1] + ur[2] * vr[2] + ur[3] * vr[3];
}

// ---------------------------------------------------------------------------
extern "C" void kernel_launch(void* const* d_in, const int* in_sizes, int n_in,
                              void* d_out, int out_size, void* d_ws, size_t ws_size,
                              hipStream_t stream)
{
    const float* U     = (const float*)d_in[0];
    const float* V     = (const float*)d_in[1];
    const float* R     = (const float*)d_in[2];
    const int*   u_idx = (const int*)d_in[3];
    const int*   v_idx = (const int*)d_in[4];

    float* out = (float*)d_out;           // out[0] = loss, out[1..] = preds
    float* ws  = (float*)d_ws;            // WAVES_TOTAL partial sums (75 KB)

    const int np = in_sizes[3];

    // Loss pass: 18750 waves, 6 waves (192 threads) per block -> 3125 blocks,
    // every wave fully populated (EXEC all-1s for WMMA).
    pmf_loss_kernel<<<WAVES_TOTAL / WAVES_PER_BLOCK, 32 * WAVES_PER_BLOCK, 0, stream>>>(
        U, V, R, ws);

    // Deterministic final reduction (+ regularization).
    pmf_reduce_kernel<<<1, REDUCE_THREADS, 0, stream>>>(ws, U, V, out);

    // Predictions.
    pmf_preds_kernel<<<(np + 255) / 256, 256, 0, stream>>>(
        U, V, u_idx, v_idx, out + 1, np);
}